// OffSetLoss_66013647339872
// MI455X (gfx1250) — compile-verified
//
#include <hip/hip_runtime.h>
#include <math.h>

#define HEADN 2
#define KK    3
#define SN    10
#define ITERS 100
#define BB    16
#define HH    128
#define WW    128
#define HW    (HH*WW)
#define NPIX  (BB*HW)      // 262144
#define TPB   256
#define NBLK  (NPIX/TPB)   // 1024 blocks

typedef float v2f __attribute__((ext_vector_type(2)));
typedef float v8f __attribute__((ext_vector_type(8)));

// ---------------------------------------------------------------------------
// Stage 1: one lane per pixel. Builds the 10x10 Gibbs kernel in VGPRs, runs
// 100 multiplicative Sinkhorn iterations (mathematically identical to the
// reference's log-domain iteration), and emits 4 partial sums per block:
//   q=0: sum(out^2)*neg   q=1: sum(o2_neg^2)*pos   q=2: pos count   q=3: loss_pos
// ---------------------------------------------------------------------------
__global__ __launch_bounds__(TPB) void offset_loss_stage1(
    const float* __restrict__ outp, const float* __restrict__ target,
    const float* __restrict__ mask, float* __restrict__ partials)
{
  const int tid = blockIdx.x * TPB + threadIdx.x;   // grid == NPIX exactly
  const int w   = tid & (WW - 1);
  const int h   = (tid >> 7) & (HH - 1);
  const int b   = tid >> 14;
  const int pix = h * WW + w;

  // ---- mask: msum, pos/neg, keep per-slot mask for loss_pos ----
  float mv[SN];
  float msum = 0.f;
  {
    const float* mp = mask + (size_t)b * SN * HW + pix;
    #pragma unroll
    for (int i = 0; i < SN; ++i) { mv[i] = mp[(size_t)i * HW]; msum += mv[i]; }
  }
  const float pos = (msum >= 1.f) ? 1.f : 0.f;
  const float neg = 1.f - pos;

  // ---- out: total sum of squares + predicted coordinates ----
  // out.reshape(B,HEAD,2,K,K,H,W); c = head*18 + axis*9 + k1*3 + k2
  // o2[axis,m] m = head*9+k1*3+k2 ; += anchor(axis? k2-1 : k1-1) + (axis? w : h)
  float o2x[SN], o2y[SN];     // m = 0..9 (used for cost)
  float sumsq = 0.f, pnsq = 0.f;
  {
    const float* op = outp + (size_t)b * (HEADN*2*KK*KK) * HW + pix;
    #pragma unroll
    for (int head = 0; head < HEADN; ++head)
      #pragma unroll
      for (int k1 = 0; k1 < KK; ++k1)
        #pragma unroll
        for (int k2 = 0; k2 < KK; ++k2) {
          const int m  = head*9 + k1*3 + k2;
          const float vx = op[(size_t)(head*18     + k1*3 + k2) * HW];
          const float vy = op[(size_t)(head*18 + 9 + k1*3 + k2) * HW];
          sumsq += vx*vx + vy*vy;
          const float ox = vx + (float)(k1 - 1) + (float)h;
          const float oy = vy + (float)(k2 - 1) + (float)w;
          if (m < SN) { o2x[m] = ox; o2y[m] = oy; }
          else        { pnsq += ox*ox + oy*oy; }
        }
  }

  // ---- target coordinates t2[i,axis] = target[b, axis*SN+i, h, w] ----
  float tx[SN], ty[SN];
  {
    const float* tp = target + (size_t)b * (2*SN) * HW + pix;
    #pragma unroll
    for (int i = 0; i < SN; ++i) {
      tx[i] = tp[(size_t)i * HW];
      ty[i] = tp[(size_t)(SN + i) * HW];
    }
  }

  // ---- cost -> scaled Gibbs kernel Kt = exp(cmin - cost), entries in (0,1] ----
  float Kt[SN][SN];
  float cmin = 1e30f;
  #pragma unroll
  for (int i = 0; i < SN; ++i)
    #pragma unroll
    for (int j = 0; j < SN; ++j) {
      const float c = fabsf(tx[i] - o2x[j]) + fabsf(ty[i] - o2y[j]);
      Kt[i][j] = c;
      cmin = fminf(cmin, c);
    }
  #pragma unroll
  for (int i = 0; i < SN; ++i)
    #pragma unroll
    for (int j = 0; j < SN; ++j)
      Kt[i][j] = __expf(cmin - Kt[i][j]);

  // ---- multiplicative Sinkhorn: a = r/(Kt b); b = c/(Kt^T a); b0 = 1 ----
  const float rmarg = 1.f / (2.f * SN);   // 0.05
  float av[SN], bv[SN];
  #pragma unroll
  for (int j = 0; j < SN; ++j) bv[j] = 1.f;
  for (int it = 0; it < ITERS; ++it) {
    #pragma unroll
    for (int i = 0; i < SN; ++i) {
      float s = 0.f;
      #pragma unroll
      for (int j = 0; j < SN; ++j) s = fmaf(Kt[i][j], bv[j], s);
      av[i] = rmarg * __builtin_amdgcn_rcpf(s);
    }
    #pragma unroll
    for (int j = 0; j < SN; ++j) {
      float s = 0.f;
      #pragma unroll
      for (int i = 0; i < SN; ++i) s = fmaf(Kt[i][j], av[i], s);
      bv[j] = rmarg * __builtin_amdgcn_rcpf(s);
    }
  }

  // ---- loss_pos: 2*SN * sum_ij mask_i * a_i * Kt_ij * b_j * cost_ij,
  //      cost recovered as cmin - log(Kt_ij); guard Kt==0 (0*inf) ----
  float lp = 0.f;
  #pragma unroll
  for (int i = 0; i < SN; ++i) {
    float row = 0.f;
    #pragma unroll
    for (int j = 0; j < SN; ++j) {
      const float kij = Kt[i][j];
      const float t   = kij * bv[j];
      const float cij = cmin - __logf(kij);
      row += (kij > 0.f) ? t * cij : 0.f;
    }
    lp = fmaf(mv[i] * av[i], row, lp);
  }
  lp *= 2.f * (float)SN;

  // ---- deterministic block reduction (shuffle tree, then LDS across waves) ----
  float p0 = sumsq * neg;
  float p1 = pnsq * pos;
  float p2 = pos;
  float p3 = lp;
  #pragma unroll
  for (int off = 16; off > 0; off >>= 1) {
    p0 += __shfl_xor(p0, off, 32);
    p1 += __shfl_xor(p1, off, 32);
    p2 += __shfl_xor(p2, off, 32);
    p3 += __shfl_xor(p3, off, 32);
  }
  __shared__ float red[4][TPB/32];
  const int wid  = threadIdx.x >> 5;
  const int lane = threadIdx.x & 31;
  if (lane == 0) { red[0][wid]=p0; red[1][wid]=p1; red[2][wid]=p2; red[3][wid]=p3; }
  __syncthreads();
  if (threadIdx.x == 0) {
    #pragma unroll
    for (int q = 0; q < 4; ++q) {
      float s = 0.f;
      #pragma unroll
      for (int ww2 = 0; ww2 < TPB/32; ++ww2) s += red[q][ww2];
      partials[q * NBLK + blockIdx.x] = s;
    }
  }
}

// ---------------------------------------------------------------------------
// Stage 2: single wave. Reduces the 4 x NBLK partials with V_WMMA_F32_16X16X4_F32
// (A = ones(16x4), B packs 64 partials per step, C accumulates column sums),
// then finishes with a 16-lane shuffle and computes the final scalar loss.
// ---------------------------------------------------------------------------
__global__ __launch_bounds__(32) void offset_loss_stage2(
    const float* __restrict__ partials, float* __restrict__ outv)
{
  const int lane = threadIdx.x;
  const v2f a_ones = {1.f, 1.f};       // A(16x4) all-ones across the wave
  float S[4];
  #pragma unroll
  for (int q = 0; q < 4; ++q) {
    v8f acc = {0.f,0.f,0.f,0.f,0.f,0.f,0.f,0.f};
    for (int chunk = 0; chunk < NBLK/64; ++chunk) {
      const float* p = partials + q * NBLK + chunk * 64;
      v2f bvals;
      bvals.x = p[lane];               // B rows K0/K2 across lanes
      bvals.y = p[lane + 32];          // B rows K1/K3 across lanes
      // D[m,n] = sum_k B[k,n] + C  -> per-column running sums (rows identical)
      acc = __builtin_amdgcn_wmma_f32_16x16x4_f32(
          false, a_ones, false, bvals, (short)0, acc, false, false);
    }
    float s = acc[0];                  // D[0,n] (n = lane%16), rows all equal
    #pragma unroll
    for (int off = 8; off > 0; off >>= 1) s += __shfl_xor(s, off, 32);
    S[q] = __shfl(s, 0, 32);
  }
  if (lane == 0) {
    const float S0 = S[0], S1 = S[1], npos = S[2], S3 = S[3];
    const float nneg = (float)NPIX - npos;
    const float loss = (sqrtf(S0) / nneg + sqrtf(S1) / npos) * 100.f
                     + S3 / (npos + 0.0001f);
    outv[0] = loss;
  }
}

extern "C" void kernel_launch(void* const* d_in, const int* in_sizes, int n_in,
                              void* d_out, int out_size, void* d_ws, size_t ws_size,
                              hipStream_t stream)
{
  const float* outp   = (const float*)d_in[0];   // (16, 36, 128, 128)
  const float* target = (const float*)d_in[1];   // (16, 20, 128, 128)
  const float* mask   = (const float*)d_in[2];   // (16, 10, 128, 128)
  float* partials = (float*)d_ws;                // 4 * NBLK floats = 16 KB
  offset_loss_stage1<<<NBLK, TPB, 0, stream>>>(outp, target, mask, partials);
  offset_loss_stage2<<<1, 32, 0, stream>>>(partials, (float*)d_out);
}